// feature_extractor_46660524704302
// MI455X (gfx1250) — compile-verified
//
#include <hip/hip_runtime.h>
#include <hip/hip_bf16.h>

typedef float v2f __attribute__((ext_vector_type(2)));
typedef float v8f __attribute__((ext_vector_type(8)));
typedef unsigned int u32x4 __attribute__((ext_vector_type(4)));
typedef int i32x8 __attribute__((ext_vector_type(8)));
typedef int i32x4 __attribute__((ext_vector_type(4)));

// ---- problem constants ----
#define NBATCH 16
#define NFRM   3747            // (480000-512)/128 + 1
#define TILES  235             // ceil(3747/16)
#define FRTOT  (NBATCH*NFRM)   // 59952
#define EPSF   1e-9f
#define BIGF   1e9f
#define TWO_PI 6.28318530717958647692f

// ---- output float offsets (tuple concatenated flat, complex64 -> re,im pairs) ----
#define OFF_FORM  0
#define OFF_EN    239808
#define OFF_CEN   299760
#define OFF_TILT  359712
#define OFF_PITCH 419664
#define OFF_VOICE 479616
#define OFF_RC    539568
#define OFF_XDS   2338128     // 59952*80 complex -> 9,592,320 floats

// ---- workspace float offsets ----
#define WS_BDS 0              // packed [256][160][2]  windowed DFT, bins 0..79 (Re cols 0..79, Im 80..159)
#define WS_BR  81920          // packed [256][272][2]  cos DFT cols 0..256 (pad->271 zero)
#define WS_BI  221184         // packed [256][272][2] -sin DFT
#define WS_D   360448         // [11][80] irfft(158) cosine matrix
#define WS_W   361328         // [512] normalized Blackman

// LDS frame tile: 16 rows, 512 floats + 2-dword pad every 256 dwords -> stride 516 (bank-conflict-free)
__device__ __forceinline__ int fidx(int m, int n) { return m * 516 + n + ((n >> 8) << 1); }

// ------------------------------------------------------------------ init: window
__global__ void fx_window(float* ws) {
    __shared__ double red[512];
    int n = threadIdx.x;
    double ph = (double)n / 511.0;
    double wr = 0.42 - 0.5 * cos(6.283185307179586 * ph) + 0.08 * cos(12.566370614359172 * ph);
    red[n] = wr * wr;
    __syncthreads();
    for (int s = 256; s; s >>= 1) { if (n < s) red[n] += red[n + s]; __syncthreads(); }
    ws[WS_W + n] = (float)(wr / sqrt(red[0]));
}

// ------------------------------------------------------------------ init: DFT bases
__global__ void fx_basis(float* ws) {
    const float* w = ws + WS_W;
    const int N1 = 512 * 160, N2 = 512 * 272;
    const int total = N1 + 2 * N2 + 11 * 80;
    for (int i = blockIdx.x * blockDim.x + threadIdx.x; i < total; i += gridDim.x * blockDim.x) {
        if (i < N1) {                                   // windowed basis for x_ds
            int k = i / 160, c = i % 160;
            float v;
            if (c < 80) { int ph = (k * c) & 511;        v =  w[k] * cosf((TWO_PI / 512.0f) * ph); }
            else        { int ph = (k * (c - 80)) & 511; v = -w[k] * sinf((TWO_PI / 512.0f) * ph); }
            ws[WS_BDS + ((k >> 1) * 160 + c) * 2 + (k & 1)] = v;
        } else if (i < N1 + N2) {                       // unwindowed cos
            int j = i - N1; int k = j / 272, c = j % 272;
            float v = (c < 257) ? cosf((TWO_PI / 512.0f) * ((k * c) & 511)) : 0.0f;
            ws[WS_BR + ((k >> 1) * 272 + c) * 2 + (k & 1)] = v;
        } else if (i < N1 + 2 * N2) {                   // unwindowed -sin
            int j = i - N1 - N2; int k = j / 272, c = j % 272;
            float v = (c < 257) ? -sinf((TWO_PI / 512.0f) * ((k * c) & 511)) : 0.0f;
            ws[WS_BI + ((k >> 1) * 272 + c) * 2 + (k & 1)] = v;
        } else {                                        // irfft(158) matrix, lags 0..10
            int j = i - N1 - 2 * N2; int m = j / 80, k = j % 80;
            float sc = (k == 0 || k == 79) ? 1.0f : 2.0f;
            int ph = (k * m) % 158;
            ws[WS_D + m * 80 + k] = sc * (1.0f / 158.0f) * cosf((TWO_PI / 158.0f) * ph);
        }
    }
}

// ------------------------------------------------------------------ Levinson-Durbin (reference-exact incl. EPS)
__device__ void fx_levinson(const float* r, int ord, float* a, float* ks) {
    float tmp[31];
    a[0] = 1.0f;
    for (int i = 1; i <= ord; ++i) a[i] = 0.0f;
    float err = r[0];
    for (int m = 1; m <= ord; ++m) {
        float acc = 0.0f;
        for (int i = 0; i < m; ++i) acc += a[i] * r[m - i];
        float k = -acc / (err + EPSF);
        if (ks) ks[m - 1] = k;
        for (int i = 0; i <= m; ++i) tmp[i] = a[i] + k * a[m - i];
        for (int i = 0; i <= m; ++i) a[i] = tmp[i];
        err *= (1.0f - k * k);
    }
}

// ------------------------------------------------------------------ main feature kernel
__global__ __launch_bounds__(256) void fx_main(const float* __restrict__ x,
                                               float* __restrict__ out,
                                               const float* __restrict__ ws) {
    __shared__ float shF[16 * 516];          // TDM destination (padded rows)
    __shared__ float shW[512];
    __shared__ float shXds[16][160];         // Re 0..79 | Im 80..159 (later reused for power)
    __shared__ float shUs[16][31];
    __shared__ float shAp[16][11];
    __shared__ float shZr[16][10], shZi[16][10];
    __shared__ float shNum[16], shDen[16], shAc0[16];
    __shared__ unsigned long long shKey[16];

    const int tid = threadIdx.x;
    const int b  = blockIdx.x / TILES;
    const int f0 = (blockIdx.x % TILES) * 16;
    const int nvalid = min(16, NFRM - f0);

    for (int n = tid; n < 512; n += 256) shW[n] = ws[WS_W + n];
    if (tid < 16) { shNum[tid] = 0.0f; shDen[tid] = 0.0f; shAc0[tid] = 0.0f; shKey[tid] = 0ULL; }

    // ---- Phase 0: Tensor Data Mover — DMA 16 overlapped frames (stride 128) into padded LDS tile ----
    if (tid < 32) {  // wave 0 issues the descriptor
        unsigned long long ga = (unsigned long long)(uintptr_t)x
                              + ((unsigned long long)b * 480000ull + (unsigned long long)f0 * 128ull) * 4ull;
        unsigned ldsOff = (unsigned)(uintptr_t)(&shF[0]);
        u32x4 g0;
        g0.x = 1u;                                            // count=1, user descriptor
        g0.y = ldsOff;                                        // lds_addr
        g0.z = (unsigned)(ga & 0xFFFFFFFFull);                // global_addr[31:0]
        g0.w = (unsigned)((ga >> 32) & 0x01FFFFFFull) | (2u << 30);  // addr[56:32] | type=2
        i32x8 g1;
        g1[0] = (int)((2u << 16) | (1u << 20) | (7u << 22) | (1u << 25)); // dsz=4B, pad_en, ivl=256dw, amt=2dw
        g1[1] = (int)((512u & 0xFFFFu) << 16);                // tensor_dim0 lo16 (=512)
        g1[2] = (int)((512u >> 16) | ((unsigned)nvalid << 16)); // tensor_dim0 hi | tensor_dim1 lo (=valid rows)
        g1[3] = (int)(512u << 16);                            // tensor_dim1 hi | tile_dim0 = 512
        g1[4] = (int)16u;                                     // tile_dim1 = 16, tile_dim2 = 0
        g1[5] = (int)128u;                                    // tensor_dim0_stride = 128 (hop)
        g1[6] = 0; g1[7] = 0;
        i32x4 z4 = {0, 0, 0, 0};
        i32x8 z8 = {0, 0, 0, 0, 0, 0, 0, 0};
        __builtin_amdgcn_tensor_load_to_lds(g0, g1, z4, z4, z8, 0);
        __builtin_amdgcn_s_wait_tensorcnt(0);
    }
    __syncthreads();

    // ---- Phase 1: WMMA f32 16x16x4 GEMMs: [16x512] x basis tiles ----
    const int wv = tid >> 5, ln = tid & 31;
    const int mrow = ln & 15, hh = ln >> 4;
    const v2f* Bds = (const v2f*)(ws + WS_BDS);
    const v2f* Br  = (const v2f*)(ws + WS_BR);
    const v2f* Bi  = (const v2f*)(ws + WS_BI);

    for (int u = wv; u < 27; u += 8) {
        if (u < 10) {                                   // x_ds GEMM (window folded into basis)
            const int col = u * 16 + mrow;
            v8f acc = {0, 0, 0, 0, 0, 0, 0, 0};
            for (int ks = 0; ks < 128; ++ks) {
                int kk = 4 * ks + 2 * hh;
                v2f a = *(const v2f*)&shF[mrow * 516 + kk + ((kk >> 8) << 1)];
                v2f bb = Bds[(2 * ks + hh) * 160 + col];
                acc = __builtin_amdgcn_wmma_f32_16x16x4_f32(false, a, false, bb, (short)0, acc, false, false);
            }
            for (int j = 0; j < 8; ++j) {
                int M = j + 8 * hh;
                shXds[M][col] = acc[j];
                if (M < nvalid) {
                    int fi = (b * NFRM + f0 + M);
                    if (col < 80) out[OFF_XDS + (fi * 80 + col) * 2]            = acc[j];
                    else          out[OFF_XDS + (fi * 80 + (col - 80)) * 2 + 1] = acc[j];
                }
            }
        } else {                                        // magnitude-spectrum GEMM pair (Re,Im)
            const int col = (u - 10) * 16 + mrow;
            v8f ar = {0, 0, 0, 0, 0, 0, 0, 0}, ai = {0, 0, 0, 0, 0, 0, 0, 0};
            for (int ks = 0; ks < 128; ++ks) {
                int kk = 4 * ks + 2 * hh;
                v2f a  = *(const v2f*)&shF[mrow * 516 + kk + ((kk >> 8) << 1)];
                v2f br2 = Br[(2 * ks + hh) * 272 + col];
                v2f bi2 = Bi[(2 * ks + hh) * 272 + col];
                ar = __builtin_amdgcn_wmma_f32_16x16x4_f32(false, a, false, br2, (short)0, ar, false, false);
                ai = __builtin_amdgcn_wmma_f32_16x16x4_f32(false, a, false, bi2, (short)0, ai, false, false);
            }
            float fb = (float)col * 31.25f;             // SR/WIN bin spacing
            for (int j = 0; j < 8; ++j) {
                int M = j + 8 * hh;
                float mag = sqrtf(ar[j] * ar[j] + ai[j] * ai[j]);  // cols>=257 are zero-basis -> 0
                atomicAdd(&shNum[M], fb * mag);
                atomicAdd(&shDen[M], mag);
            }
        }
    }

    // ---- Phase 2: time-domain autocorrelations (cheaper than FFT for <=321 lags) ----
    for (int tt = tid; tt < 16 * 31; tt += 256) {       // circular autocorr of windowed frame, lags 0..30
        int f = tt / 31, l = tt % 31;
        float s = 0.0f;
        for (int n = 0; n < 512; ++n) {
            int n2 = (n + l) & 511;
            s += (shF[fidx(f, n)] * shW[n]) * (shF[fidx(f, n2)] * shW[n2]);
        }
        shUs[f][l] = s;
    }
    for (int tt = tid; tt < 16 * 290; tt += 256) {      // linear autocorr, lag 0 and 32..320
        int f = tt / 290, li = tt % 290;
        int lag = li ? li + 31 : 0;
        float s = 0.0f;
        for (int n = 0; n < 512 - lag; ++n) s += shF[fidx(f, n)] * shF[fidx(f, n + lag)];
        if (lag == 0) shAc0[f] = s;
        else {
            unsigned ub = __float_as_uint(s);
            unsigned mono = (ub & 0x80000000u) ? ~ub : (ub | 0x80000000u);  // order-preserving
            unsigned long long key = ((unsigned long long)mono << 32)
                                   | (unsigned long long)(0xFFFFFFFFu - (unsigned)lag); // ties -> smallest lag
            atomicMax(&shKey[f], key);
        }
    }
    __syncthreads();

    // ---- Phase 3: per-frame scalar features (one lane per frame) ----
    if (tid < 16) {
        const int f = tid;
        const bool valid = f < nvalid;
        const int fi = b * NFRM + f0 + f;
        // power of ds bins (reuse shXds row as |X|^2)
        for (int k = 0; k < 80; ++k) {
            float re = shXds[f][k], im = shXds[f][80 + k];
            shXds[f][k] = re * re + im * im;
        }
        float dsac[11];
        const float* D = ws + WS_D;
        for (int m = 0; m <= 10; ++m) {
            float s = 0.0f;
            for (int k = 0; k < 80; ++k) s += D[m * 80 + k] * shXds[f][k];
            dsac[m] = s;
        }
        float tilt = dsac[1] / (dsac[0] + EPSF);
        float ap[11];
        fx_levinson(dsac, 10, ap, nullptr);
        for (int i = 0; i <= 10; ++i) shAp[f][i] = ap[i];
        float a30[31], ks30[30];
        fx_levinson(&shUs[f][0], 30, a30, ks30);

        float ac0 = shAc0[f];
        unsigned long long key = shKey[f];
        unsigned mono = (unsigned)(key >> 32);
        unsigned ub = (mono & 0x80000000u) ? (mono & 0x7FFFFFFFu) : ~mono;
        float peak = __uint_as_float(ub);
        int lag = (int)(0xFFFFFFFFu - (unsigned)(key & 0xFFFFFFFFull));
        float vo = peak / (ac0 + EPSF);
        float voicing = (vo > 0.3f) ? 1.0f : 0.0f;
        float pitch = 16000.0f / (float)lag * voicing;

        if (valid) {
            out[OFF_EN    + fi] = 10.0f * log10f(ac0 + EPSF);
            out[OFF_CEN   + fi] = shNum[f] / (shDen[f] + EPSF);
            out[OFF_TILT  + fi] = tilt;
            out[OFF_PITCH + fi] = pitch;
            out[OFF_VOICE + fi] = voicing;
            for (int m = 0; m < 30; ++m) out[OFF_RC + fi * 30 + m] = ks30[m];
        }
    }
    __syncthreads();

    // ---- Phase 4: Durand-Kerner (lane = frame x root), roots exchanged via LDS ----
    const int df = tid / 10, dr = tid % 10;
    if (tid < 160) {
        float zr = 1.0f, zi = 0.0f;
        for (int i = 0; i <= dr; ++i) { float t = 0.4f * zr - 0.9f * zi; zi = 0.9f * zr + 0.4f * zi; zr = t; }
        shZr[df][dr] = zr; shZi[df][dr] = zi;
    }
    __syncthreads();
    for (int it = 0; it < 30; ++it) {
        float nzr = 0.0f, nzi = 0.0f;
        if (tid < 160) {
            float zr = shZr[df][dr], zi = shZi[df][dr];
            float pr = shAp[df][0], pi = 0.0f;
            for (int i = 1; i <= 10; ++i) {
                float t = pr * zr - pi * zi + shAp[df][i];
                pi = pr * zi + pi * zr; pr = t;
            }
            float drr = 1.0f, dri = 0.0f;
            for (int j = 0; j < 10; ++j) {
                if (j == dr) continue;
                float er = zr - shZr[df][j], ei = zi - shZi[df][j];
                float t = drr * er - dri * ei; dri = drr * ei + dri * er; drr = t;
            }
            float den = drr * drr + dri * dri;
            nzr = zr - (pr * drr + pi * dri) / den;
            nzi = zi - (pi * drr - pr * dri) / den;
        }
        __syncthreads();
        if (tid < 160) { shZr[df][dr] = nzr; shZi[df][dr] = nzi; }
        __syncthreads();
    }

    // ---- Phase 5: formants ----
    if (tid < 16 && tid < nvalid) {
        const int fi = b * NFRM + f0 + tid;
        float fr[10];
        for (int r = 0; r < 10; ++r) {
            float zr = shZr[tid][r], zi = shZi[tid][r];
            float ang = atan2f(zi, zr);
            float fq = ang * (5000.0f / TWO_PI);
            fr[r] = (zi > 0.0f && fq > 0.0f) ? fq : BIGF;
        }
        for (int i = 1; i < 10; ++i) {                  // insertion sort ascending
            float v = fr[i]; int j = i - 1;
            while (j >= 0 && fr[j] > v) { fr[j + 1] = fr[j]; --j; }
            fr[j + 1] = v;
        }
        for (int q = 0; q < 4; ++q) out[OFF_FORM + fi * 4 + q] = (fr[q] < BIGF) ? fr[q] : 0.0f;
    }
}

// ------------------------------------------------------------------ launch
extern "C" void kernel_launch(void* const* d_in, const int* in_sizes, int n_in,
                              void* d_out, int out_size, void* d_ws, size_t ws_size,
                              hipStream_t stream) {
    const float* x = (const float*)d_in[0];
    float* out = (float*)d_out;
    float* ws  = (float*)d_ws;   // needs ~1.45 MB for DFT bases + window + irfft matrix
    (void)in_sizes; (void)n_in; (void)out_size; (void)ws_size;

    fx_window<<<1, 512, 0, stream>>>(ws);
    fx_basis<<<708, 512, 0, stream>>>(ws);
    fx_main<<<NBATCH * TILES, 256, 0, stream>>>(x, out, ws);
}